// CombinedModel_cossim_70437463655096
// MI455X (gfx1250) — compile-verified
//
#include <hip/hip_runtime.h>

// ---------------------------------------------------------------------------
// CombinedModel cosine-sim pipeline for MI455X (gfx1250, wave32, WMMA).
//   B=8, N=4096, M=2048, D=512, T=2 (row 1 of text used).
//   out[b,m,0:512]   = (sum_n exp(sim[n,m]) * features[b,n,:]) / sum_n exp(sim[n,m])
//   out[b,m,512:1024] = text[1,m,:]
// All matrix math uses V_WMMA_F32_16X16X4_F32 (fp32-exact vs reference).
// Kernel 3 stages feature tiles with GLOBAL_LOAD_ASYNC_TO_LDS_B128 (ASYNCcnt)
// overlapped with the sim WMMA phase, and feeds B-fragments with
// DS_LOAD_2ADDR_STRIDE64_B32 (two K-rows, 2048B apart, in one DS op).
// ---------------------------------------------------------------------------

typedef float v2f __attribute__((ext_vector_type(2)));
typedef float v8f __attribute__((ext_vector_type(8)));

#define EPS_NORM 1e-8f

#define Bdim 8
#define Ndim 4096
#define Mdim 2048
#define Ddim 512

__device__ __forceinline__ v8f wmma_f32(v2f a, v2f b, v8f c) {
  // 8 args: (neg_a, A, neg_b, B, c_mod, C, reuse_a, reuse_b)
  return __builtin_amdgcn_wmma_f32_16x16x4_f32(false, a, false, b, (short)0, c,
                                               false, false);
}

// Async DMA: 16 bytes/lane, global -> LDS, tracked by ASYNCcnt.
// Per CDNA5 ISA: the same INST_OFFSET is added to BOTH the
// global and the LDS per-lane address, so one immediate advances the copy.
template <int OFF>
__device__ __forceinline__ void async_copy_b128(unsigned lds_off,
                                                const void* gaddr) {
  asm volatile("global_load_async_to_lds_b128 %0, %1, off offset:%c2"
               :
               : "v"(lds_off), "v"(gaddr), "i"(OFF)
               : "memory");
}

__device__ __forceinline__ void wait_asynccnt0() {
  asm volatile("s_wait_asynccnt 0x0" ::: "memory");
}

__device__ __forceinline__ void wait_dscnt0() {
  asm volatile("s_wait_dscnt 0x0" ::: "memory");
}

// One DS op returns the floats at LDS byte addresses base and base+2048: the (K, K+1) pair of one
// column of a row-major 512-float-stride tile (stride64 => offset*256B).
__device__ __forceinline__ v2f ds_pair_load(unsigned base) {
  v2f r;
  asm volatile("ds_load_2addr_stride64_b32 %0, %1 offset0:0 offset1:8"
               : "=v"(r)
               : "v"(base)
               : "memory");
  return r;
}

// ---------------------------------------------------------------------------
// Kernel 0: pair-interleave W:  Wp[k>>1][d][2] = { W[k][d], W[k+1][d] }
// so a WMMA B-fragment (K-pair for one column) is one contiguous b64.
// ---------------------------------------------------------------------------
__global__ __launch_bounds__(256) void k_packW(const float* __restrict__ W,
                                               float* __restrict__ Wp) {
  const int k2 = blockIdx.x;        // 0..255 (row pair)
  const int c  = threadIdx.x * 2;   // 0..510 (2 cols per thread)
  const float* r0 = W + (size_t)(2 * k2) * Ddim;
  const float* r1 = r0 + Ddim;
  v2f a0 = *(const v2f*)(r0 + c);
  v2f a1 = *(const v2f*)(r1 + c);
  float4 s = {a0.x, a1.x, a0.y, a1.y};
  *(float4*)(Wp + ((size_t)k2 * Ddim + c) * 2) = s;
}

// ---------------------------------------------------------------------------
// Kernel 1: fn = normalize_rows(features @ W + b)
// Block = 8 waves; block handles 16 rows (of B*N) x all 512 cols.
// Wave w owns columns [w*64, w*64+64) as 4 WMMA 16x16 C-tiles, K=512.
// ---------------------------------------------------------------------------
__global__ __launch_bounds__(256) void k_linear_norm(
    const float* __restrict__ F,     // [B*N, 512]
    const float* __restrict__ Wp,    // [256, 512, 2] pair-interleaved W
    const float* __restrict__ bias,  // [512]
    float* __restrict__ fn)          // [B*N, 512] (ws)
{
  __shared__ float rowsum[16];

  const int wave = threadIdx.x >> 5;
  const int lane = threadIdx.x & 31;
  const int hi   = lane >> 4;   // 0/1 : K-half for A/B frags
  const int lo   = lane & 15;
  const int row0 = blockIdx.x * 16;   // over flat B*N

  if (threadIdx.x < 16) rowsum[threadIdx.x] = 0.0f;
  __syncthreads();

  v8f zero = {};
  v8f acc[4] = {zero, zero, zero, zero};

  const float* arow  = F + (size_t)(row0 + lo) * Ddim;
  const int    dbase = wave * 64 + lo;

  for (int k = 0; k < Ddim; k += 4) {
    const int koff = k + 2 * hi;
    const int p    = koff >> 1;                      // K-pair index
    v2f a = *(const v2f*)(arow + koff);              // A[m=lo][koff..koff+1]
#pragma unroll
    for (int t = 0; t < 4; ++t) {
      const int dcol = dbase + t * 16;
      v2f bf = *(const v2f*)(Wp + ((size_t)p * Ddim + dcol) * 2);  // b64
      acc[t] = wmma_f32(a, bf, acc[t]);
    }
  }

  // bias add (bias depends only on column = lane/tile)
#pragma unroll
  for (int t = 0; t < 4; ++t) {
    const float bv = bias[dbase + t * 16];
#pragma unroll
    for (int v = 0; v < 8; ++v) acc[t][v] += bv;
  }

  // row sum-of-squares: C layout -> lane holds (row = v+8*hi, col = dbase+t*16)
  float s[8];
#pragma unroll
  for (int v = 0; v < 8; ++v) {
    float x = 0.0f;
#pragma unroll
    for (int t = 0; t < 4; ++t) x += acc[t][v] * acc[t][v];
    s[v] = x;
  }
#pragma unroll
  for (int mask = 1; mask <= 8; mask <<= 1) {
#pragma unroll
    for (int v = 0; v < 8; ++v) s[v] += __shfl_xor(s[v], mask);
  }
  if (lo == 0) {
#pragma unroll
    for (int v = 0; v < 8; ++v) atomicAdd(&rowsum[v + 8 * hi], s[v]);
  }
  __syncthreads();

#pragma unroll
  for (int v = 0; v < 8; ++v) {
    const int   r  = v + 8 * hi;
    const float sc = 1.0f / fmaxf(sqrtf(rowsum[r]), EPS_NORM);
    float* orow = fn + (size_t)(row0 + r) * Ddim;
#pragma unroll
    for (int t = 0; t < 4; ++t) orow[dbase + t * 16] = acc[t][v] * sc;
  }
}

// ---------------------------------------------------------------------------
// Kernel 2: tn = normalize_rows(text[1])   (one wave per 512-float row)
// ---------------------------------------------------------------------------
__global__ __launch_bounds__(256) void k_tnorm(const float* __restrict__ t1,
                                               float* __restrict__ tn) {
  const int wave = threadIdx.x >> 5;
  const int lane = threadIdx.x & 31;
  const int row  = blockIdx.x * 8 + wave;

  const float* src = t1 + (size_t)row * Ddim + lane * 16;
  float4 d[4];
  float  ss = 0.0f;
#pragma unroll
  for (int j = 0; j < 4; ++j) {
    d[j] = *(const float4*)(src + j * 4);
    ss += d[j].x * d[j].x + d[j].y * d[j].y + d[j].z * d[j].z + d[j].w * d[j].w;
  }
#pragma unroll
  for (int mask = 16; mask >= 1; mask >>= 1) ss += __shfl_xor(ss, mask);
  const float sc = 1.0f / fmaxf(sqrtf(ss), EPS_NORM);

  float* dst = tn + (size_t)row * Ddim + lane * 16;
#pragma unroll
  for (int j = 0; j < 4; ++j) {
    float4 o = d[j];
    o.x *= sc; o.y *= sc; o.z *= sc; o.w *= sc;
    *(float4*)(dst + j * 4) = o;
  }
}

// ---------------------------------------------------------------------------
// Kernel 3: fused  sim -> exp -> streaming weighted sum (flash-attn style).
// Block = 8 waves, one (b, 16-wide m-tile). Streams n in tiles of 16:
//  Stage:   async DMA features n-tile -> LDS (row-major), overlapped with A
//  Phase A: split-K sim GEMM (each wave: K-slice of 64; tn frags hoisted
//           into registers -- they are n-loop invariant) -> partials in LDS
//  Phase B: 256-lane parallel reduce + exp -> e_t[m][n] in LDS, denom regs
//  Phase C: acc[m, d] += e^T @ F-tile  (B-frags: one ds_load_2addr_stride64
//           per (K-pair, column); batched, then one s_wait_dscnt 0)
// Finally: out[b,m,d] = acc/denom ; out[b,m,512+d] = t1[m,d]
// ---------------------------------------------------------------------------
__global__ __launch_bounds__(256) void k_fused(
    const float* __restrict__ F,    // features [B*N, 512]
    const float* __restrict__ fn,   // normalized f1 [B*N, 512]
    const float* __restrict__ tn,   // normalized t1 [M, 512]
    const float* __restrict__ t1,   // raw text[1]   [M, 512]
    float* __restrict__ out)        // [B, M, 1024]
{
  __shared__ float fstage[16 * Ddim];     // 32 KB row-major features tile
  __shared__ float partials[8 * 256];     // per-wave partial 16x16 sim tiles
  __shared__ float e_t[256];              // exp(sim) transposed: [m][n]
  __shared__ float denomAcc[16];

  const int wave = threadIdx.x >> 5;
  const int lane = threadIdx.x & 31;
  const int hi   = lane >> 4;
  const int lo   = lane & 15;
  const int b    = blockIdx.x >> 7;          // M/16 = 128 tiles per batch
  const int m0   = (blockIdx.x & 127) * 16;

  if (threadIdx.x < 16) denomAcc[threadIdx.x] = 0.0f;

  v8f zero = {};
  v8f acc[4] = {zero, zero, zero, zero};
  float denomReg = 0.0f;

  const int idx = wave * 32 + lane;  // 0..255 over the 16x16 tile
  const int nl  = idx >> 4;          // local n of this lane's e element
  const int ml  = idx & 15;          // local m of this lane's e element

  const float* fnb   = fn + (size_t)b * Ndim * Ddim;
  const float* Fb    = F + (size_t)b * Ndim * Ddim;
  const int    ks    = wave * 64;        // this wave's K-slice for sim
  const int    dbase = wave * 64 + lo;   // this wave's d-columns for numer

  // tn B-fragments: invariant over the whole n-loop -> keep in 32 VGPRs.
  const float* brow = tn + (size_t)(m0 + lo) * Ddim;
  v2f bfragT[16];
#pragma unroll
  for (int kk4 = 0; kk4 < 16; ++kk4)
    bfragT[kk4] = *(const v2f*)(brow + ks + kk4 * 4 + 2 * hi);

  // Loop-invariant LDS addresses (wave-relative low 32 bits of flat addr):
  const unsigned fsbase   = (unsigned)(size_t)&fstage[0];
  const unsigned ldsStage = fsbase + threadIdx.x * 16;  // this lane's DMA slot
  unsigned cb[4][4];  // Phase-C B-frag bases: [(k4*4+2*hi)*512 + dcol] * 4B
#pragma unroll
  for (int k4 = 0; k4 < 4; ++k4) {
#pragma unroll
    for (int t = 0; t < 4; ++t)
      cb[k4][t] = fsbase + (unsigned)(((k4 * 4 + 2 * hi) * Ddim + dbase + t * 16) * 4);
  }

  __syncthreads();

  for (int n0 = 0; n0 < Ndim; n0 += 16) {
    // ---- Async-stage the 16x512 features tile (32 KB) into LDS.
    // 256 lanes x 16B x 8 slabs; each instruction is a fully-coalesced 4KB.
    {
      const char* g = (const char*)(Fb + (size_t)n0 * Ddim) + threadIdx.x * 16;
      async_copy_b128<0>(ldsStage, g);
      async_copy_b128<4096>(ldsStage, g);
      async_copy_b128<8192>(ldsStage, g);
      async_copy_b128<12288>(ldsStage, g);
      async_copy_b128<16384>(ldsStage, g);
      async_copy_b128<20480>(ldsStage, g);
      async_copy_b128<24576>(ldsStage, g);
      async_copy_b128<28672>(ldsStage, g);
    }

    // ---- Phase A: partial sim over this wave's K slice (overlaps the DMA).
    v8f sim = zero;
    const float* arow = fnb + (size_t)(n0 + lo) * Ddim;
#pragma unroll
    for (int kk4 = 0; kk4 < 16; ++kk4) {
      const int koff = ks + kk4 * 4 + 2 * hi;
      v2f a = *(const v2f*)(arow + koff);   // A[n=lo][k..k+1]
      sim = wmma_f32(a, bfragT[kk4], sim);
    }
#pragma unroll
    for (int v = 0; v < 8; ++v)
      partials[wave * 256 + (v + 8 * hi) * 16 + lo] = sim[v];

    wait_asynccnt0();   // our slice of the DMA has landed in LDS
    __syncthreads();    // everyone's slices visible

    // ---- Phase B: cross-wave reduce + exp --------------------------------
    float sv = 0.0f;
#pragma unroll
    for (int p = 0; p < 8; ++p) sv += partials[p * 256 + idx];
    const float ev = __expf(sv);   // sim in [-1,1]: no max-subtraction needed
    denomReg += ev;                // this lane always owns column ml
    e_t[ml * 16 + nl] = ev;
    __syncthreads();

    // ---- Phase C: acc[m,d] += e^T (16x16) @ F-tile (from LDS) ------------
    v2f af[4], bf[4][4];
#pragma unroll
    for (int k4 = 0; k4 < 4; ++k4) {
      af[k4] = *(const v2f*)(&e_t[lo * 16 + k4 * 4 + 2 * hi]);
#pragma unroll
      for (int t = 0; t < 4; ++t) bf[k4][t] = ds_pair_load(cb[k4][t]);
    }
    wait_dscnt0();      // drain our untracked DS pair-loads
#pragma unroll
    for (int k4 = 0; k4 < 4; ++k4) {
#pragma unroll
      for (int t = 0; t < 4; ++t) acc[t] = wmma_f32(af[k4], bf[k4][t], acc[t]);
    }
    // fstage is overwritten by the next iteration's DMA: fence it.
    __syncthreads();
  }

  atomicAdd(&denomAcc[ml], denomReg);
  __syncthreads();

#pragma unroll
  for (int v = 0; v < 8; ++v) {
    const int   m   = v + 8 * hi;                // C layout row = our m
    const float inv = 1.0f / denomAcc[m];
    float*       orow = out + (size_t)(b * Mdim + m0 + m) * (2 * Ddim);
    const float* trow = t1 + (size_t)(m0 + m) * Ddim;
#pragma unroll
    for (int t = 0; t < 4; ++t) {
      const int d = dbase + t * 16;
      orow[d]        = acc[t][v] * inv;  // fm = numer / denom
      orow[Ddim + d] = trow[d];          // concat: broadcast text[1]
    }
  }
}

// ---------------------------------------------------------------------------
extern "C" void kernel_launch(void* const* d_in, const int* in_sizes, int n_in,
                              void* d_out, int out_size, void* d_ws,
                              size_t ws_size, hipStream_t stream) {
  const float* features = (const float*)d_in[0];  // [8,4096,512]
  const float* text     = (const float*)d_in[1];  // [2,2048,512]
  const float* W        = (const float*)d_in[2];  // [512,512]
  const float* bias     = (const float*)d_in[3];  // [512]
  float*       out      = (float*)d_out;          // [8,2048,1024]

  float* fn = (float*)d_ws;                              // 64 MB
  float* tn = fn + (size_t)Bdim * Ndim * Ddim;           // +4 MB
  float* Wp = tn + (size_t)Mdim * Ddim;                  // +1 MB
  const float* t1 = text + (size_t)1 * Mdim * Ddim;      // text[1]

  k_packW<<<Ddim / 2, 256, 0, stream>>>(W, Wp);
  k_linear_norm<<<(Bdim * Ndim) / 16, 256, 0, stream>>>(features, Wp, bias, fn);
  k_tnorm<<<Mdim / 8, 256, 0, stream>>>(t1, tn);
  k_fused<<<Bdim * (Mdim / 16), 256, 0, stream>>>(features, fn, tn, t1, out);
}